// ChamferLoss_69148973465956
// MI455X (gfx1250) — compile-verified
//
#include <hip/hip_runtime.h>

typedef __attribute__((ext_vector_type(2))) float v2f;
typedef __attribute__((ext_vector_type(8))) float v8f;

#define M_ 8
#define N_ 4096
#define D_ 64
#define NT (N_ / 16)  // 256 16-wide tiles per batch

__device__ __forceinline__ v8f wmma_f32(const v2f& a, const v2f& b, const v8f& c) {
  return __builtin_amdgcn_wmma_f32_16x16x4_f32(
      /*neg_a=*/false, a, /*neg_b=*/false, b,
      /*c_mod=*/(short)0, c, /*reuse_a=*/false, /*reuse_b=*/false);
}

// Load one 16x64 f32 tile in WMMA-operand layout: lane lh owns point lh,
// chunk kk supplies K = 4*kk + 2*hi (+0,+1). base already includes lh*D + 2*hi.
__device__ __forceinline__ void load_tile(v2f (&b)[16], const float* __restrict__ base) {
#pragma unroll
  for (int kk = 0; kk < 16; ++kk) b[kk] = *(const v2f*)(base + 4 * kk);
}

// ---------------------------------------------------------------------------
// Kernel 0: initialize column-min array to +inf.
// ---------------------------------------------------------------------------
__global__ void chamfer_init(unsigned* __restrict__ colMinBits) {
  int idx = blockIdx.x * blockDim.x + threadIdx.x;
  if (idx < M_ * N_) colMinBits[idx] = 0x7F800000u;
}

// ---------------------------------------------------------------------------
// Kernel 1: one wave per (batch, 16-row tile). Double-buffered B tiles,
// two independent WMMA accumulation chains per tile.
// ---------------------------------------------------------------------------
__global__ __launch_bounds__(32) void chamfer_tiles(
    const float* __restrict__ A, const float* __restrict__ B,
    float* __restrict__ rowPartial, unsigned* __restrict__ colMinBits) {
  const int wg   = blockIdx.x;
  const int m    = wg / NT;       // batch
  const int it   = wg % NT;       // row tile index
  const int lane = threadIdx.x;   // 0..31
  const int lh   = lane & 15;
  const int hi   = lane >> 4;

  const float INF = __uint_as_float(0x7F800000u);

  // ---- A tile (16x64) resident in registers for the whole j-loop ----------
  const float* Ab = A + ((size_t)m * N_ + (size_t)it * 16) * D_ + lh * D_ + 2 * hi;
  v2f a[16];
  load_tile(a, Ab);
  float a2p = 0.f;
#pragma unroll
  for (int kk = 0; kk < 16; ++kk) a2p += a[kk].x * a[kk].x + a[kk].y * a[kk].y;
  float a2row = a2p + __shfl_xor(a2p, 16);   // |a|^2 of row lh
  float a2acc[8];
#pragma unroll
  for (int v = 0; v < 8; ++v) a2acc[v] = __shfl(a2row, v + 8 * hi);

  float rowMin[8];
#pragma unroll
  for (int v = 0; v < 8; ++v) rowMin[v] = INF;

  const float* Bb = B + (size_t)m * N_ * D_ + lh * D_ + 2 * hi;

  // ---- per-tile processing: |b|^2, 2 interleaved WMMA chains, min updates --
  auto process = [&](const v2f(&b)[16], int j) {
    float b2p = 0.f;
#pragma unroll
    for (int kk = 0; kk < 16; ++kk) b2p += b[kk].x * b[kk].x + b[kk].y * b[kk].y;
    float b2col = b2p + __shfl_xor(b2p, 16);  // |b|^2 of col j*16 + lh

    v8f accE = {}, accO = {};
#pragma unroll
    for (int kk = 0; kk < 8; ++kk) {
      accE = wmma_f32(a[2 * kk],     b[2 * kk],     accE);
      accO = wmma_f32(a[2 * kk + 1], b[2 * kk + 1], accO);
    }

    float colMin = INF;
#pragma unroll
    for (int v = 0; v < 8; ++v) {
      float ab = accE[v] + accO[v];
      float d  = fmaxf(a2acc[v] + b2col - 2.0f * ab, 0.0f);
      rowMin[v] = fminf(rowMin[v], d);
      colMin    = fminf(colMin, d);
    }
    colMin = fminf(colMin, __shfl_xor(colMin, 16));
    if (hi == 0) {
      // dist >= 0 -> IEEE bits order-preserving -> deterministic float min.
      atomicMin(&colMinBits[m * N_ + j * 16 + lh], __float_as_uint(colMin));
    }
  };

  // ---- software-pipelined j-loop: load tile j+1 while computing tile j -----
  v2f b0[16], b1[16];
  load_tile(b0, Bb);  // tile 0
  for (int j = 0; j < NT; j += 2) {
    load_tile(b1, Bb + (size_t)(j + 1) * 16 * D_);
    process(b0, j);
    int jn = (j + 2 < NT) ? (j + 2) : (NT - 1);  // clamp (redundant last load)
    load_tile(b0, Bb + (size_t)jn * 16 * D_);
    process(b1, j + 1);
  }

  // ---- row-min reduction across the 16 lanes of each half ------------------
#pragma unroll
  for (int v = 0; v < 8; ++v) {
#pragma unroll
    for (int mask = 1; mask < 16; mask <<= 1)
      rowMin[v] = fminf(rowMin[v], __shfl_xor(rowMin[v], mask));
  }
  float s = 0.f;
#pragma unroll
  for (int v = 0; v < 8; ++v) s += rowMin[v];  // rows hi*8 .. hi*8+7
  s += __shfl_xor(s, 16);                      // all 16 rows of this tile
  if (lane == 0) rowPartial[wg] = s;           // deterministic partial
}

// ---------------------------------------------------------------------------
// Kernel 2: per batch, fixed-order sum of 4096 col-mins + 256 row partials.
// ---------------------------------------------------------------------------
__global__ void chamfer_final(const unsigned* __restrict__ colMinBits,
                              const float* __restrict__ rowPartial,
                              float* __restrict__ out) {
  __shared__ float sm[256];
  const int m = blockIdx.x, t = threadIdx.x;
  float s = 0.f;
  for (int k = t; k < N_; k += 256) s += __uint_as_float(colMinBits[m * N_ + k]);
  for (int r = t; r < NT; r += 256) s += rowPartial[m * NT + r];
  sm[t] = s;
  __syncthreads();
  for (int off = 128; off > 0; off >>= 1) {
    if (t < off) sm[t] += sm[t + off];
    __syncthreads();
  }
  if (t == 0) out[m] = sm[0];
}

// ---------------------------------------------------------------------------
extern "C" void kernel_launch(void* const* d_in, const int* in_sizes, int n_in,
                              void* d_out, int out_size, void* d_ws,
                              size_t ws_size, hipStream_t stream) {
  const float* predicted = (const float*)d_in[0];  // [8,4096,64] f32
  const float* expected  = (const float*)d_in[1];  // [8,4096,64] f32
  float* out = (float*)d_out;                      // [8] f32

  // Workspace layout: colMinBits (8*4096 u32) | rowPartial (8*256 f32)
  unsigned* colMinBits = (unsigned*)d_ws;
  float* rowPartial = (float*)((char*)d_ws + (size_t)M_ * N_ * sizeof(unsigned));

  chamfer_init<<<(M_ * N_ + 255) / 256, 256, 0, stream>>>(colMinBits);
  chamfer_tiles<<<M_ * NT, 32, 0, stream>>>(predicted, expected, rowPartial,
                                            colMinBits);
  chamfer_final<<<M_, 256, 0, stream>>>(colMinBits, rowPartial, out);
}